// BoxSampler_34926674051111
// MI455X (gfx1250) — compile-verified
//
#include <hip/hip_runtime.h>

#define B1 50000
#define B2 512
#define NPOS 128
#define ROWS_PER_BLOCK 128   // 8 waves * 16 rows
#define KPT 49               // ceil(50000/1024)

typedef float v2f __attribute__((ext_vector_type(2)));
typedef float v8f __attribute__((ext_vector_type(8)));
typedef int   v4i_vec __attribute__((vector_size(16)));   // matches builtin proto

#ifndef __has_builtin
#define __has_builtin(x) 0
#endif

#if __has_builtin(__builtin_amdgcn_global_load_async_to_lds_b128)
#define USE_ASYNC_LDS 1
#else
#define USE_ASYNC_LDS 0
#endif

#if __has_builtin(__builtin_amdgcn_wmma_f32_16x16x4_f32)
#define USE_WMMA 1
#else
#define USE_WMMA 0
#endif

// ---------------------------------------------------------------- init
__global__ void init_kernel(unsigned long long* tbest, int* neg_cnt) {
  int i = threadIdx.x;
  if (i < B2) tbest[i] = 0ull;
  if (i == B2) *neg_cnt = 0;
}

// ---------------------------------------------------------------- IoU
// Each wave computes a 16(proposal-rows) x 512(targets) strip as 32 tiles of
// 16x16.  The union denominator outer-sum (area1+area2) comes from
// V_WMMA_F32_16X16X4_F32; intersection + division + running row-argmax are
// done directly in the WMMA C/D register layout (lane = column N, VGPR k =
// row k + 8*(lane>=16)).
__global__ __launch_bounds__(256) void iou_kernel(
    const float* __restrict__ pb, const float* __restrict__ tb,
    float* __restrict__ max_iou, int* __restrict__ input_idx,
    unsigned long long* __restrict__ tbest) {
  __shared__ float s_raw[B2 * 4];
  __shared__ float s_tx0[B2], s_ty0[B2], s_tx1[B2], s_ty1[B2], s_ta[B2];

  const int tid  = threadIdx.x;
  const int lane = tid & 31;
  const int wv   = tid >> 5;

  // ---- stage raw target boxes (8KB) into LDS ----
#if USE_ASYNC_LDS
  {
    const float* g0 = tb + tid * 4;
    float*       l0 = &s_raw[tid * 4];
    __builtin_amdgcn_global_load_async_to_lds_b128(
        (__attribute__((address_space(1))) v4i_vec*)g0,
        (__attribute__((address_space(3))) v4i_vec*)l0, 0, 0);
    __builtin_amdgcn_global_load_async_to_lds_b128(
        (__attribute__((address_space(1))) v4i_vec*)(g0 + 1024),
        (__attribute__((address_space(3))) v4i_vec*)(l0 + 1024), 0, 0);
#if __has_builtin(__builtin_amdgcn_s_wait_asynccnt)
    __builtin_amdgcn_s_wait_asynccnt(0);
#else
    asm volatile("s_wait_asynccnt 0" ::: "memory");
#endif
  }
#else
  for (int i = tid; i < B2 * 4; i += 256) s_raw[i] = tb[i];
#endif
  __syncthreads();

  // convert to corner form + area
  for (int j = tid; j < B2; j += 256) {
    float xc = s_raw[4 * j + 0], yc = s_raw[4 * j + 1];
    float w  = s_raw[4 * j + 2], h  = s_raw[4 * j + 3];
    s_tx0[j] = xc - 0.5f * w; s_ty0[j] = yc - 0.5f * h;
    s_tx1[j] = xc + 0.5f * w; s_ty1[j] = yc + 0.5f * h;
    s_ta[j]  = w * h;
  }
  __syncthreads();

  const int waveBase = blockIdx.x * ROWS_PER_BLOCK + wv * 16;
  const int half     = lane >> 4;            // 0: rows 0..7, 1: rows 8..15
  const int rbase    = waveBase + half * 8;

  // this lane's 8 proposal rows (corner form)
  float px0[8], py0[8], px1[8], py1[8], pa[8];
  const float4* pb4 = (const float4*)pb;
#pragma unroll
  for (int k = 0; k < 8; ++k) {
    int row = rbase + k;
    if (row < B1) {
      float4 b = pb4[row];
      px0[k] = b.x - 0.5f * b.z; py0[k] = b.y - 0.5f * b.w;
      px1[k] = b.x + 0.5f * b.z; py1[k] = b.y + 0.5f * b.w;
      pa[k]  = b.z * b.w;
    } else {
      px0[k] = py0[k] = px1[k] = py1[k] = 0.0f; pa[k] = 0.0f;
    }
  }

  // A fragment (16x4 f32): lanes 0-15 hold row M=lane, (K0,K1)=(area1, 1);
  // lanes 16-31 hold (K2,K3)=(0,0).
  v2f afrag;
  if (lane < 16) {
    int row = waveBase + lane;
    float a = 0.0f;
    if (row < B1) { float4 b = pb4[row]; a = b.z * b.w; }
    afrag.x = a; afrag.y = 1.0f;
  } else {
    afrag.x = 0.0f; afrag.y = 0.0f;
  }

  float rMax[8]; int rArg[8];
#pragma unroll
  for (int k = 0; k < 8; ++k) { rMax[k] = -1.0f; rArg[k] = 0; }

  for (int t = 0; t < B2 / 16; ++t) {
    const int   jcol = t * 16 + (lane & 15);
    const float tx0 = s_tx0[jcol], ty0 = s_ty0[jcol];
    const float tx1 = s_tx1[jcol], ty1 = s_ty1[jcol];
    const float ta  = s_ta[jcol];

    // B fragment (4x16 f32): lanes 0-15 col N=lane, (K0,K1)=(1, area2);
    // lanes 16-31 (K2,K3)=(0,0).
    v2f bfrag;
    if (lane < 16) { bfrag.x = 1.0f; bfrag.y = ta; }
    else           { bfrag.x = 0.0f; bfrag.y = 0.0f; }

    v8f S;
#if USE_WMMA
    {
      v8f cz = {};
      // D[M][N] = area1[M]*1 + 1*area2[N]  (exact fp32, single rounding)
      S = __builtin_amdgcn_wmma_f32_16x16x4_f32(false, afrag, false, bfrag,
                                                (short)0, cz, false, false);
    }
#else
#pragma unroll
    for (int k = 0; k < 8; ++k) S[k] = pa[k] + ta;
#endif

    float colBest = -1.0f; int colRow = 0;
#pragma unroll
    for (int k = 0; k < 8; ++k) {
      float ix = fminf(px1[k], tx1) - fmaxf(px0[k], tx0);
      float iy = fminf(py1[k], ty1) - fmaxf(py0[k], ty0);
      ix = fmaxf(ix, 0.0f); iy = fmaxf(iy, 0.0f);
      float inter = ix * iy;
      float iou   = inter / (S[k] - inter);
      if (iou > rMax[k]) { rMax[k] = iou; rArg[k] = jcol; }  // first-index tie
      int grow = rbase + k;
      if (grow < B1 && iou > colBest) { colBest = iou; colRow = grow; }
    }
    // merge rows 0-7 / 8-15 halves for this column
    float ov = __shfl_xor(colBest, 16, 32);
    int   orr = __shfl_xor(colRow, 16, 32);
    if (ov > colBest || (ov == colBest && orr < colRow)) { colBest = ov; colRow = orr; }
    if (lane < 16 && colBest >= 0.0f) {
      // iou>=0 so float bits are monotone; ~row makes ties pick smallest row
      unsigned long long key =
          ((unsigned long long)__float_as_uint(colBest) << 32) |
          (unsigned)(~(unsigned)colRow);
      atomicMax(&tbest[jcol], key);
    }
  }

  // reduce row max/argmax across the 16 columns (stays inside each half)
#pragma unroll
  for (int m = 1; m <= 8; m <<= 1) {
#pragma unroll
    for (int k = 0; k < 8; ++k) {
      float ov = __shfl_xor(rMax[k], m, 32);
      int   oa = __shfl_xor(rArg[k], m, 32);
      if (ov > rMax[k] || (ov == rMax[k] && oa < rArg[k])) { rMax[k] = ov; rArg[k] = oa; }
    }
  }
  if ((lane & 15) == 0) {
#pragma unroll
    for (int k = 0; k < 8; ++k) {
      int grow = rbase + k;
      if (grow < B1) { max_iou[grow] = rMax[k]; input_idx[grow] = rArg[k]; }
    }
  }
}

// ---------------------------------------------------------------- masks
__global__ void mask_kernel(const float* __restrict__ max_iou,
                            unsigned char* __restrict__ pos,
                            unsigned char* __restrict__ neg) {
  int i = blockIdx.x * blockDim.x + threadIdx.x;
  if (i < B1) {
    float m = max_iou[i];
    pos[i] = (m > 0.7f) ? 1 : 0;
    neg[i] = (m < 0.3f) ? 1 : 0;
  }
}

__global__ void scatter_kernel(const unsigned long long* __restrict__ tbest,
                               unsigned char* __restrict__ pos,
                               unsigned char* __restrict__ neg) {
  int j = threadIdx.x;
  if (j < B2) {
    int row = (int)(~(unsigned)(tbest[j] & 0xFFFFFFFFull));
    if (row >= 0 && row < B1) { pos[row] = 1; neg[row] = 0; }
  }
}

__global__ void count_kernel(const unsigned char* __restrict__ neg, int* cnt) {
  __shared__ int s;
  if (threadIdx.x == 0) s = 0;
  __syncthreads();
  int i = blockIdx.x * blockDim.x + threadIdx.x;
  int v = (i < B1) ? (int)neg[i] : 0;
  if (v) atomicAdd(&s, 1);
  __syncthreads();
  if (threadIdx.x == 0 && s) atomicAdd(cnt, s);
}

// ---------------------------------------------------------------- top-k
// block 0: positives -> out[0..127]=pos_input_idx, out[128..255]=pos_target_idx
// block 1: negatives -> out[256..383]=neg_input_idx
__global__ __launch_bounds__(1024) void topk_kernel(
    const float* __restrict__ pos_noise, const float* __restrict__ neg_noise,
    const unsigned char* __restrict__ pos_mask,
    const unsigned char* __restrict__ neg_mask,
    const int* __restrict__ input_idx, const int* __restrict__ neg_cnt,
    float* __restrict__ out) {
  __shared__ float sV[1024];
  __shared__ int   sI[1024];
  __shared__ int   sWin;
  const int  tid   = threadIdx.x;
  const bool isPos = (blockIdx.x == 0);
  const bool negFB = (*neg_cnt == 0);

  float sc[KPT];
#pragma unroll
  for (int k = 0; k < KPT; ++k) {
    int i = tid + k * 1024;
    float s = -3.0f;                        // out of range
    if (i < B1) {
      if (isPos) {
        s = pos_mask[i] ? pos_noise[i] : -1.0f;
      } else {
        int m = negFB ? (pos_mask[i] ? 0 : 1) : (int)neg_mask[i];
        s = m ? neg_noise[i] : -1.0f;
      }
    }
    sc[k] = s;
  }

  for (int r = 0; r < NPOS; ++r) {
    float bv = -4.0f; int bi = 0x7fffffff;
#pragma unroll
    for (int k = 0; k < KPT; ++k) {         // strict > keeps smallest index
      if (sc[k] > bv) { bv = sc[k]; bi = tid + k * 1024; }
    }
    sV[tid] = bv; sI[tid] = bi;
    __syncthreads();
    for (int st = 512; st > 0; st >>= 1) {
      if (tid < st) {
        float ov = sV[tid + st]; int oi = sI[tid + st];
        if (ov > sV[tid] || (ov == sV[tid] && oi < sI[tid])) { sV[tid] = ov; sI[tid] = oi; }
      }
      __syncthreads();
    }
    if (tid == 0) {
      int win = sI[0];
      sWin = win;
      if (isPos) {
        out[r]        = (float)win;
        out[NPOS + r] = (float)input_idx[win];
      } else {
        out[2 * NPOS + r] = (float)win;
      }
    }
    __syncthreads();
    int win = sWin;
    if ((win & 1023) == tid) sc[win >> 10] = -2.0f;   // mark taken
    __syncthreads();
  }
}

// ---------------------------------------------------------------- launch
extern "C" void kernel_launch(void* const* d_in, const int* in_sizes, int n_in,
                              void* d_out, int out_size, void* d_ws,
                              size_t ws_size, hipStream_t stream) {
  (void)in_sizes; (void)n_in; (void)out_size; (void)ws_size;
  const float* pb = (const float*)d_in[0];  // (1,50000,4)
  const float* tb = (const float*)d_in[1];  // (1,512,4)
  const float* pn = (const float*)d_in[2];  // (50000,)
  const float* nn = (const float*)d_in[3];  // (50000,)

  char* ws = (char*)d_ws;
  float*              max_iou   = (float*)(ws + 0);        // 200000 B
  int*                input_idx = (int*)(ws + 200704);     // 200000 B
  unsigned long long* tbest     = (unsigned long long*)(ws + 401408); // 4 KB
  unsigned char*      pos_mask  = (unsigned char*)(ws + 405504);      // 50 KB
  unsigned char*      neg_mask  = (unsigned char*)(ws + 455680);      // 50 KB
  int*                neg_cnt   = (int*)(ws + 505856);
  float*              out       = (float*)d_out;

  init_kernel<<<1, 1024, 0, stream>>>(tbest, neg_cnt);
  iou_kernel<<<(B1 + ROWS_PER_BLOCK - 1) / ROWS_PER_BLOCK, 256, 0, stream>>>(
      pb, tb, max_iou, input_idx, tbest);
  mask_kernel<<<(B1 + 255) / 256, 256, 0, stream>>>(max_iou, pos_mask, neg_mask);
  scatter_kernel<<<1, B2, 0, stream>>>(tbest, pos_mask, neg_mask);
  count_kernel<<<(B1 + 255) / 256, 256, 0, stream>>>(neg_mask, neg_cnt);
  topk_kernel<<<2, 1024, 0, stream>>>(pn, nn, pos_mask, neg_mask, input_idx,
                                      neg_cnt, out);
}